// LinkPredictor_79233556677240
// MI455X (gfx1250) — compile-verified
//
#include <hip/hip_runtime.h>

typedef float v2f __attribute__((ext_vector_type(2)));
typedef float v8f __attribute__((ext_vector_type(8)));

#define N_NODES 50000
#define N_EDGES 800000
#define IN_CH   128
#define HID     64

// ---------------------------------------------------------------------------
// Degree / normalization
// ---------------------------------------------------------------------------
__global__ void k_init_deg(float* __restrict__ deg) {
    int i = blockIdx.x * blockDim.x + threadIdx.x;
    if (i < N_NODES) deg[i] = 1.0f;                 // self-loop contribution
}

__global__ void k_count_deg(const int* __restrict__ dst, float* __restrict__ deg) {
    int e = blockIdx.x * blockDim.x + threadIdx.x;
    if (e < N_EDGES) atomicAdd(&deg[dst[e]], 1.0f);
}

__global__ void k_deg_to_dinv(float* __restrict__ deg) {
    int i = blockIdx.x * blockDim.x + threadIdx.x;
    if (i < N_NODES) deg[i] = rsqrtf(deg[i]);       // deg >= 1 always
}

// ---------------------------------------------------------------------------
// GEMM: out[N_NODES x HID] = (RELU? max(A,0):A)[N_NODES x K] @ W[K x HID]
// One wave per 16-row M-tile; fp32 WMMA 16x16x4; 4 N-tiles (HID=64).
// A 16x4 f32 layout: lanes 0-15 = M rows; lane>>4 selects K pair {0,1}/{2,3};
// VGPR0 = first K of pair, VGPR1 = second. B mirrors with lane&15 = N column.
// C/D: VGPR r holds row (r + 8*(lane>>4)), col (lane&15).
// ---------------------------------------------------------------------------
template <int K, bool RELU>
__global__ void k_gemm_wmma(const float* __restrict__ A,
                            const float* __restrict__ W,
                            float* __restrict__ out) {
    const int lane  = threadIdx.x & 31;
    const int wave  = threadIdx.x >> 5;
    const int mtile = blockIdx.x * (blockDim.x >> 5) + wave;
    if (mtile * 16 >= N_NODES) return;              // wave-uniform: EXEC stays full

    const int laneN = lane & 15;
    const int kHalf = lane >> 4;

    const float* __restrict__ arow = A + (mtile * 16 + laneN) * (long)K;

    v8f acc0 = {}, acc1 = {}, acc2 = {}, acc3 = {};

    for (int k0 = 0; k0 < K; k0 += 4) {
        const int ka = k0 + 2 * kHalf;
        v2f a;
        a.x = arow[ka];
        a.y = arow[ka + 1];
        if (RELU) { a.x = fmaxf(a.x, 0.0f); a.y = fmaxf(a.y, 0.0f); }

        const float* __restrict__ w0 = W + ka * HID;
        const float* __restrict__ w1 = w0 + HID;

        v2f b;
        b.x = w0[laneN];      b.y = w1[laneN];
        acc0 = __builtin_amdgcn_wmma_f32_16x16x4_f32(false, a, false, b,
                                                     (short)0, acc0, false, false);
        b.x = w0[16 + laneN]; b.y = w1[16 + laneN];
        acc1 = __builtin_amdgcn_wmma_f32_16x16x4_f32(false, a, false, b,
                                                     (short)0, acc1, false, false);
        b.x = w0[32 + laneN]; b.y = w1[32 + laneN];
        acc2 = __builtin_amdgcn_wmma_f32_16x16x4_f32(false, a, false, b,
                                                     (short)0, acc2, false, false);
        b.x = w0[48 + laneN]; b.y = w1[48 + laneN];
        acc3 = __builtin_amdgcn_wmma_f32_16x16x4_f32(false, a, false, b,
                                                     (short)0, acc3, false, false);
    }

    const int mBase = mtile * 16 + 8 * kHalf;
#pragma unroll
    for (int r = 0; r < 8; ++r) {
        float* __restrict__ orow = out + (long)(mBase + r) * HID;
        orow[laneN]      = acc0[r];
        orow[16 + laneN] = acc1[r];
        orow[32 + laneN] = acc2[r];
        orow[48 + laneN] = acc3[r];
    }
}

// ---------------------------------------------------------------------------
// z[i,c] = h[i,c] * dinv[i]^2 + bias[c]   (self-loop term + bias)
// ---------------------------------------------------------------------------
__global__ void k_init_z(const float* __restrict__ h, const float* __restrict__ dinv,
                         const float* __restrict__ bias, float* __restrict__ z) {
    int t = blockIdx.x * blockDim.x + threadIdx.x;
    if (t < N_NODES * HID) {
        int i = t >> 6;
        int c = t & 63;
        float d = dinv[i];
        z[t] = h[t] * d * d + bias[c];
    }
}

// ---------------------------------------------------------------------------
// z[dst] += h[src] * dinv[src]*dinv[dst]   — 16 lanes per edge, float4 each
// ---------------------------------------------------------------------------
__global__ void k_scatter(const int* __restrict__ src, const int* __restrict__ dst,
                          const float* __restrict__ dinv,
                          const float* __restrict__ h, float* __restrict__ z) {
    int t = blockIdx.x * blockDim.x + threadIdx.x;
    int e = t >> 4;
    if (e >= N_EDGES) return;
    int c4 = (t & 15) * 4;
    int s = src[e], d = dst[e];
    float norm = dinv[s] * dinv[d];
    const float4 hv = *(const float4*)(h + (long)s * HID + c4);
    float* zp = z + (long)d * HID + c4;
    atomicAdd(zp + 0, hv.x * norm);
    atomicAdd(zp + 1, hv.y * norm);
    atomicAdd(zp + 2, hv.z * norm);
    atomicAdd(zp + 3, hv.w * norm);
}

// ---------------------------------------------------------------------------
// out[e] = dot(z[src_e], z[dst_e]) over 64 channels — 16 lanes per edge
// ---------------------------------------------------------------------------
__global__ void k_decode(const int* __restrict__ src, const int* __restrict__ dst,
                         const float* __restrict__ z, float* __restrict__ out) {
    int t = blockIdx.x * blockDim.x + threadIdx.x;
    int e = t >> 4;
    if (e >= N_EDGES) return;
    int c4 = (t & 15) * 4;
    int s = src[e], d = dst[e];
    float4 a = *(const float4*)(z + (long)s * HID + c4);
    float4 b = *(const float4*)(z + (long)d * HID + c4);
    float p = a.x * b.x + a.y * b.y + a.z * b.z + a.w * b.w;
    p += __shfl_xor(p, 1, 32);
    p += __shfl_xor(p, 2, 32);
    p += __shfl_xor(p, 4, 32);
    p += __shfl_xor(p, 8, 32);
    if ((t & 15) == 0) out[e] = p;
}

// ---------------------------------------------------------------------------
extern "C" void kernel_launch(void* const* d_in, const int* in_sizes, int n_in,
                              void* d_out, int out_size, void* d_ws, size_t ws_size,
                              hipStream_t stream) {
    const float* x   = (const float*)d_in[0];
    const int*   ei  = (const int*)d_in[1];
    const float* W1  = (const float*)d_in[2];
    const float* b1  = (const float*)d_in[3];
    const float* W2  = (const float*)d_in[4];
    const float* b2  = (const float*)d_in[5];
    float*       out = (float*)d_out;

    const int* src = ei;             // edge_index[0]
    const int* dst = ei + N_EDGES;   // edge_index[1]

    // Workspace layout (floats): dinv | bufA (h1 / t) | bufB (z1 / z2)
    float* dinv = (float*)d_ws;
    float* bufA = dinv + 50048;
    float* bufB = bufA + (long)N_NODES * HID;

    const int B = 256;

    // 1) symmetric-normalization coefficients
    k_init_deg<<<(N_NODES + B - 1) / B, B, 0, stream>>>(dinv);
    k_count_deg<<<(N_EDGES + B - 1) / B, B, 0, stream>>>(dst, dinv);
    k_deg_to_dinv<<<(N_NODES + B - 1) / B, B, 0, stream>>>(dinv);

    // 2) layer 1: h1 = x @ W1 ; z1 = A_hat h1 + b1
    const int mtiles = N_NODES / 16;                 // 3125, exact
    const int wavesPerBlock = B / 32;                // 8
    const int gemmBlocks = (mtiles + wavesPerBlock - 1) / wavesPerBlock;
    k_gemm_wmma<IN_CH, false><<<gemmBlocks, B, 0, stream>>>(x, W1, bufA);
    k_init_z<<<(N_NODES * HID + B - 1) / B, B, 0, stream>>>(bufA, dinv, b1, bufB);
    k_scatter<<<(N_EDGES * 16 + B - 1) / B, B, 0, stream>>>(src, dst, dinv, bufA, bufB);

    // 3) layer 2: t = relu(z1) @ W2 ; z2 = A_hat t + b2
    k_gemm_wmma<HID, true><<<gemmBlocks, B, 0, stream>>>(bufB, W2, bufA);
    k_init_z<<<(N_NODES * HID + B - 1) / B, B, 0, stream>>>(bufA, dinv, b2, bufB);
    // note: k_init_z overwrites bufB only after gemm2 consumed it (stream-ordered)
    k_scatter<<<(N_EDGES * 16 + B - 1) / B, B, 0, stream>>>(src, dst, dinv, bufA, bufB);

    // 4) decode
    k_decode<<<(N_EDGES * 16 + B - 1) / B, B, 0, stream>>>(src, dst, bufB, out);
}